// NeuralSNLDS_65051574665614
// MI455X (gfx1250) — compile-verified
//
#include <hip/hip_runtime.h>
#include <math.h>

#define B_DIM 64
#define T_DIM 2048
#define LDIM  16
#define HDIM  64
#define KST   8
#define LOG2PI_F 1.8378770664093453f

typedef float v2f __attribute__((ext_vector_type(2)));
typedef float v8f __attribute__((ext_vector_type(8)));

// D = A(16x4) * B(4x16) + C, f32, wave32 WMMA (CDNA5)
__device__ __forceinline__ v8f wmma4(v2f a, v2f b, v8f c) {
    return __builtin_amdgcn_wmma_f32_16x16x4_f32(false, a, false, b, (short)0, c,
                                                 false, false);
}

// fast stable softplus: max(x,0) + log(1 + exp(-|x|)) using HW exp/log.
// exp(-|x|) in (0,1], so 1+e in (1,2] where v_log_f32 is well conditioned.
__device__ __forceinline__ float softplus_fast(float x) {
    return fmaxf(x, 0.0f) + __logf(1.0f + __expf(-fabsf(x)));
}

// ---------------------------------------------------------------------------
// prep: Cholesky terms for covs/init_cov, log-softmax of pi and Q rows.
// One block of 32 threads; perf-irrelevant.
// ---------------------------------------------------------------------------
__global__ void prep_kernel(const float* __restrict__ pi,
                            const float* __restrict__ Q,
                            const float* __restrict__ init_cov,
                            const float* __restrict__ covs,
                            float* __restrict__ Linv0, float* __restrict__ LinvT,
                            float* __restrict__ ld0,   float* __restrict__ ldT,
                            float* __restrict__ logpi, float* __restrict__ Qlog) {
    const int tid = threadIdx.x;
    if (tid < 16) {
        const float* C    = (tid < 8) ? (covs + tid * 256) : (init_cov + (tid - 8) * 256);
        float*       Linv = (tid < 8) ? (LinvT + tid * 256) : (Linv0 + (tid - 8) * 256);
        float*       ldh  = (tid < 8) ? (ldT + tid) : (ld0 + (tid - 8));
        float S[16][16];
        for (int r = 0; r < 16; ++r)
            for (int c = 0; c < 16; ++c) {
                float s = (r == c) ? 1e-6f : 0.0f;
                for (int d = 0; d < 16; ++d) s += C[r * 16 + d] * C[c * 16 + d];
                S[r][c] = s;
            }
        // in-place lower Cholesky
        float logdet = 0.0f;
        for (int j = 0; j < 16; ++j) {
            float d = S[j][j];
            for (int p = 0; p < j; ++p) d -= S[j][p] * S[j][p];
            d = sqrtf(d);
            S[j][j] = d;
            logdet += logf(d);
            float inv = 1.0f / d;
            for (int r = j + 1; r < 16; ++r) {
                float v = S[r][j];
                for (int p = 0; p < j; ++p) v -= S[r][p] * S[j][p];
                S[r][j] = v * inv;
            }
        }
        *ldh = logdet;
        // invert lower-triangular L column by column
        for (int c = 0; c < 16; ++c) {
            float x[16];
            for (int r = 0; r < 16; ++r) x[r] = 0.0f;
            x[c] = 1.0f / S[c][c];
            for (int r = c + 1; r < 16; ++r) {
                float a = 0.0f;
                for (int p = c; p < r; ++p) a += S[r][p] * x[p];
                x[r] = -a / S[r][r];
            }
            for (int r = 0; r < 16; ++r) Linv[r * 16 + c] = (r >= c) ? x[r] : 0.0f;
        }
    } else if (tid == 16) {
        float m = pi[0];
        for (int i = 1; i < KST; ++i) m = fmaxf(m, pi[i]);
        float s = 0.0f;
        for (int i = 0; i < KST; ++i) s += expf(pi[i] - m);
        float l = m + logf(s);
        for (int i = 0; i < KST; ++i) logpi[i] = pi[i] - l;
    } else if (tid >= 17 && tid < 17 + KST) {
        const int r = tid - 17;
        float m = Q[r * KST];
        for (int j = 1; j < KST; ++j) m = fmaxf(m, Q[r * KST + j]);
        float s = 0.0f;
        for (int j = 0; j < KST; ++j) s += expf(Q[r * KST + j] - m);
        float l = m + logf(s);
        for (int j = 0; j < KST; ++j) Qlog[r * KST + j] = Q[r * KST + j] - l;
    }
}

// ---------------------------------------------------------------------------
// emis0: le[b,0,k] from init distribution (tiny, scalar)
// ---------------------------------------------------------------------------
__global__ void emis0_kernel(const float* __restrict__ z,
                             const float* __restrict__ init_mean,
                             const float* __restrict__ Linv0,
                             const float* __restrict__ ld0,
                             float* __restrict__ le) {
    const int g = blockIdx.x * blockDim.x + threadIdx.x;
    if (g >= B_DIM * KST) return;
    const int b = g >> 3, k = g & 7;
    float diff[16];
    for (int d = 0; d < 16; ++d)
        diff[d] = z[(size_t)b * T_DIM * LDIM + d] - init_mean[k * LDIM + d];
    float maha = 0.0f;
    for (int r = 0; r < 16; ++r) {
        float y = 0.0f;
        for (int c = 0; c <= r; ++c) y += Linv0[(k * 16 + r) * 16 + c] * diff[c];
        maha += y * y;
    }
    le[(size_t)b * T_DIM * KST + k] = -0.5f * maha - ld0[k] - 8.0f * LOG2PI_F;
}

// ---------------------------------------------------------------------------
// emis: MLP emission log-probs for t = 1..T-1, WMMA f32 16x16x4.
// Block = (batch, 16-timestep tile); wave w handles state k = w.
// ---------------------------------------------------------------------------
#define HSTRIDE 68  // padded row stride for per-wave 16x64 LDS tile (bank-safe)
#define ZSTRIDE 17  // padded row stride for 17x16 z tile

__global__ __launch_bounds__(256) void emis_kernel(
    const float* __restrict__ z, const float* __restrict__ W1,
    const float* __restrict__ b1, const float* __restrict__ W2,
    const float* __restrict__ b2, const float* __restrict__ LinvT,
    const float* __restrict__ ldT, float* __restrict__ le) {
    const int b     = blockIdx.x >> 7;    // /128 tiles
    const int tile  = blockIdx.x & 127;
    const int t0    = tile * 16;          // zp rows t0..t0+15 -> outputs t0+1..t0+16
    const int tid   = threadIdx.x;
    const int k     = tid >> 5;           // wave id == state
    const int lane  = tid & 31;
    const int nsub  = lane & 15;
    const int khalf = lane >> 4;

    __shared__ float zt[17 * ZSTRIDE];
    __shared__ float hbuf[8][16 * HSTRIDE];
    float* hw = hbuf[k];

    // cooperative load of 17 rows of z (rows t0..t0+16, clamped)
    for (int idx = tid; idx < 17 * 16; idx += 256) {
        int r = idx >> 4, d = idx & 15;
        int t = t0 + r;
        if (t > T_DIM - 1) t = T_DIM - 1;
        zt[r * ZSTRIDE + d] = z[((size_t)b * T_DIM + t) * LDIM + d];
    }
    __syncthreads();

    // ---- A fragments of Z (reused across all N tiles of GEMM1) ----
    v2f az[4];
#pragma unroll
    for (int kc = 0; kc < 4; ++kc) {
        az[kc].x = zt[nsub * ZSTRIDE + 4 * kc + 2 * khalf + 0];
        az[kc].y = zt[nsub * ZSTRIDE + 4 * kc + 2 * khalf + 1];
    }

    // ---- GEMM1: h(16x64) = Z(16x16) @ W1[k]^T(16x64), + bias, softplus ----
#pragma unroll
    for (int nt = 0; nt < 4; ++nt) {
        v8f acc;
#pragma unroll
        for (int v = 0; v < 8; ++v) acc[v] = 0.0f;
        const int hrow = nt * 16 + nsub;  // W1 row index (h dim)
#pragma unroll
        for (int kc = 0; kc < 4; ++kc) {
            v2f bw;
            bw.x = W1[((size_t)k * HDIM + hrow) * LDIM + 4 * kc + 2 * khalf + 0];
            bw.y = W1[((size_t)k * HDIM + hrow) * LDIM + 4 * kc + 2 * khalf + 1];
            acc = wmma4(az[kc], bw, acc);
        }
        const float bias = b1[k * HDIM + hrow];
#pragma unroll
        for (int v = 0; v < 8; ++v) {
            float sp = softplus_fast(acc[v] + bias);
            int   m  = v + khalf * 8;
            hw[m * HSTRIDE + hrow] = sp;
        }
    }

    // ---- GEMM2: mu(16x16) = h(16x64) @ W2[k]^T(64x16) ----
    v8f mu;
#pragma unroll
    for (int v = 0; v < 8; ++v) mu[v] = 0.0f;
#pragma unroll
    for (int kc = 0; kc < 16; ++kc) {
        v2f ah, bw;
        ah.x = hw[nsub * HSTRIDE + 4 * kc + 2 * khalf + 0];
        ah.y = hw[nsub * HSTRIDE + 4 * kc + 2 * khalf + 1];
        bw.x = W2[((size_t)k * LDIM + nsub) * HDIM + 4 * kc + 2 * khalf + 0];
        bw.y = W2[((size_t)k * LDIM + nsub) * HDIM + 4 * kc + 2 * khalf + 1];
        mu = wmma4(ah, bw, mu);
    }

    // ---- diff = z_next - (mu + b2), stage to LDS for re-layout ----
    const float b2v = b2[k * LDIM + nsub];
#pragma unroll
    for (int v = 0; v < 8; ++v) {
        int m = v + khalf * 8;
        hw[m * HSTRIDE + nsub] = zt[(m + 1) * ZSTRIDE + nsub] - (mu[v] + b2v);
    }

    // ---- GEMM3: y(16x16) = diff(16x16) @ Linv[k]^T(16x16) ----
    v8f y;
#pragma unroll
    for (int v = 0; v < 8; ++v) y[v] = 0.0f;
#pragma unroll
    for (int kc = 0; kc < 4; ++kc) {
        v2f ad, bl;
        ad.x = hw[nsub * HSTRIDE + 4 * kc + 2 * khalf + 0];
        ad.y = hw[nsub * HSTRIDE + 4 * kc + 2 * khalf + 1];
        bl.x = LinvT[(k * 16 + nsub) * 16 + 4 * kc + 2 * khalf + 0];
        bl.y = LinvT[(k * 16 + nsub) * 16 + 4 * kc + 2 * khalf + 1];
        y = wmma4(ad, bl, y);
    }

    // ---- maha row-reduce + store le ----
    const float ldk = ldT[k];
#pragma unroll
    for (int v = 0; v < 8; ++v) {
        float s = y[v] * y[v];
        s += __shfl_xor(s, 1, 32);
        s += __shfl_xor(s, 2, 32);
        s += __shfl_xor(s, 4, 32);
        s += __shfl_xor(s, 8, 32);
        int m    = v + khalf * 8;
        int tout = t0 + 1 + m;
        if (nsub == 0 && tout < T_DIM)
            le[((size_t)b * T_DIM + tout) * KST + k] = -0.5f * s - ldk - 8.0f * LOG2PI_F;
    }
}

// ---------------------------------------------------------------------------
// fwd: sequential forward scan. lane = (b, i); K-reductions within 8-lane group
// ---------------------------------------------------------------------------
__global__ __launch_bounds__(256) void fwd_kernel(const float* __restrict__ le,
                                                  const float* __restrict__ logpi,
                                                  const float* __restrict__ Qlog,
                                                  float* __restrict__ la,
                                                  float* __restrict__ logZ) {
    const int g = blockIdx.x * 256 + threadIdx.x;
    const int b = g >> 3, i = g & 7;
    float Qcol[8];
#pragma unroll
    for (int j = 0; j < 8; ++j) Qcol[j] = Qlog[j * KST + i];  // QT[i][j]

    const float* leb = le + (size_t)b * T_DIM * KST;
    float*       lab = la + (size_t)b * T_DIM * KST;
    float*       lzb = logZ + (size_t)b * T_DIM;

    // t = 0
    float lp = leb[i] + logpi[i];
    float m  = lp;
    m = fmaxf(m, __shfl_xor(m, 1, 32));
    m = fmaxf(m, __shfl_xor(m, 2, 32));
    m = fmaxf(m, __shfl_xor(m, 4, 32));
    float e = __expf(lp - m);
    e += __shfl_xor(e, 1, 32);
    e += __shfl_xor(e, 2, 32);
    e += __shfl_xor(e, 4, 32);
    float lz = m + __logf(e);
    float av = lp - lz;
    lab[i] = av;
    if (i == 0) lzb[0] = lz;

    for (int t = 1; t < T_DIM; ++t) {
        const float let = leb[t * KST + i];
        float vj[8];
#pragma unroll
        for (int j = 0; j < 8; ++j) vj[j] = Qcol[j] + __shfl(av, j, 8);
        float mm = vj[0];
#pragma unroll
        for (int j = 1; j < 8; ++j) mm = fmaxf(mm, vj[j]);
        float s = 0.0f;
#pragma unroll
        for (int j = 0; j < 8; ++j) s += __expf(vj[j] - mm);
        float lpi = let + mm + __logf(s);
        float m2  = lpi;
        m2 = fmaxf(m2, __shfl_xor(m2, 1, 32));
        m2 = fmaxf(m2, __shfl_xor(m2, 2, 32));
        m2 = fmaxf(m2, __shfl_xor(m2, 4, 32));
        float e2 = __expf(lpi - m2);
        e2 += __shfl_xor(e2, 1, 32);
        e2 += __shfl_xor(e2, 2, 32);
        e2 += __shfl_xor(e2, 4, 32);
        float lz2 = m2 + __logf(e2);
        av        = lpi - lz2;
        lab[t * KST + i] = av;
        if (i == 0) lzb[t] = lz2;
    }
}

// ---------------------------------------------------------------------------
// bwd: reverse scan; fuses gamma = exp(la+lb) and paired stores (float4 x2).
// ---------------------------------------------------------------------------
__global__ __launch_bounds__(256) void bwd_kernel(const float* __restrict__ le,
                                                  const float* __restrict__ la,
                                                  const float* __restrict__ Qlog,
                                                  const float* __restrict__ logZ,
                                                  float* __restrict__ gamma,
                                                  float* __restrict__ paired) {
    const int g = blockIdx.x * 256 + threadIdx.x;
    const int b = g >> 3, i = g & 7;
    float Qrow[8];
#pragma unroll
    for (int j = 0; j < 8; ++j) Qrow[j] = Qlog[i * KST + j];

    const float* leb = le + (size_t)b * T_DIM * KST;
    const float* lab = la + (size_t)b * T_DIM * KST;
    const float* lzb = logZ + (size_t)b * T_DIM;
    float*       gb  = gamma + (size_t)b * T_DIM * KST;
    float*       pb  = paired + (size_t)b * (T_DIM - 1) * KST * KST;

    float lb = 0.0f;
    gb[(T_DIM - 1) * KST + i] = __expf(lab[(T_DIM - 1) * KST + i]);

    for (int t = T_DIM - 2; t >= 0; --t) {
        const float u   = lb + leb[(t + 1) * KST + i];  // lb[t+1,i] + le[t+1,i]
        const float lz1 = lzb[t + 1];
        const float lat = lab[t * KST + i];
        float aij[8];
        float mm = -3.4e38f;
#pragma unroll
        for (int j = 0; j < 8; ++j) {
            aij[j] = Qrow[j] + __shfl(u, j, 8);
            mm     = fmaxf(mm, aij[j]);
        }
        float s = 0.0f;
#pragma unroll
        for (int j = 0; j < 8; ++j) s += __expf(aij[j] - mm);
        const float lbn = mm + __logf(s) - lz1;
        gb[t * KST + i] = __expf(lat + lbn);

        // paired[b,t,i,0..7] : two 16-byte stores (base is 32B aligned per lane)
        const float c = lat - lz1;
        float4 p0, p1;
        p0.x = __expf(aij[0] + c);
        p0.y = __expf(aij[1] + c);
        p0.z = __expf(aij[2] + c);
        p0.w = __expf(aij[3] + c);
        p1.x = __expf(aij[4] + c);
        p1.y = __expf(aij[5] + c);
        p1.z = __expf(aij[6] + c);
        p1.w = __expf(aij[7] + c);
        float4* pr = (float4*)(pb + ((size_t)t * KST + i) * KST);
        pr[0] = p0;
        pr[1] = p1;

        lb = lbn;
    }
}

// ---------------------------------------------------------------------------
extern "C" void kernel_launch(void* const* d_in, const int* in_sizes, int n_in,
                              void* d_out, int out_size, void* d_ws, size_t ws_size,
                              hipStream_t stream) {
    const float* z         = (const float*)d_in[0];
    const float* pi        = (const float*)d_in[1];
    const float* Q         = (const float*)d_in[2];
    const float* init_mean = (const float*)d_in[3];
    const float* init_cov  = (const float*)d_in[4];
    const float* covs      = (const float*)d_in[5];
    const float* W1        = (const float*)d_in[6];
    const float* b1        = (const float*)d_in[7];
    const float* W2        = (const float*)d_in[8];
    const float* b2        = (const float*)d_in[9];

    float* gamma  = (float*)d_out;                                    // B*T*K
    float* paired = gamma + (size_t)B_DIM * T_DIM * KST;              // B*(T-1)*K*K
    float* logZ   = paired + (size_t)B_DIM * (T_DIM - 1) * KST * KST; // B*T

    float* ws    = (float*)d_ws;
    float* le    = ws;                                  // B*T*K
    float* la    = le + (size_t)B_DIM * T_DIM * KST;    // B*T*K
    float* Linv0 = la + (size_t)B_DIM * T_DIM * KST;    // K*256
    float* LinvT = Linv0 + KST * 256;                   // K*256
    float* ld0   = LinvT + KST * 256;                   // K
    float* ldT   = ld0 + KST;                           // K
    float* logpi = ldT + KST;                           // K
    float* Qlog  = logpi + KST;                         // K*K

    prep_kernel<<<1, 32, 0, stream>>>(pi, Q, init_cov, covs, Linv0, LinvT, ld0,
                                      ldT, logpi, Qlog);
    emis0_kernel<<<2, 256, 0, stream>>>(z, init_mean, Linv0, ld0, le);
    emis_kernel<<<B_DIM * 128, 256, 0, stream>>>(z, W1, b1, W2, b2, LinvT, ldT, le);
    fwd_kernel<<<2, 256, 0, stream>>>(le, logpi, Qlog, la, logZ);
    bwd_kernel<<<2, 256, 0, stream>>>(le, la, Qlog, logZ, gamma, paired);
}